// custom_loss_36206574305815
// MI455X (gfx1250) — compile-verified
//
#include <hip/hip_runtime.h>
#include <math.h>

// ---------------------------------------------------------------------------
// Reference reduces to: sqrt( sum( (output - target)^2 ) )   (Frobenius norm)
// Proof: strict sign mismatch  => (o-t)^2 == (|o|+|t|)^2
//        signs agree / zeros   => (o-t)^2 == (|o|-|t|)^2
// Bandwidth-bound: 268 MB streamed once -> ~11.5 us floor @ 23.3 TB/s.
// Strategy: counted + unrolled loop of b128 non-temporal loads (8 loads in
// flight per wave), packed v_pk_fma_f32 accumulation, wave32 reduction via
// V_WMMA_F32_16X16X4_F32 (ones-matrix trick, full f32), LDS cross-wave,
// two-pass block-partial reduction through d_ws.
// ---------------------------------------------------------------------------

typedef float v2f __attribute__((ext_vector_type(2)));
typedef float v4f __attribute__((ext_vector_type(4)));
typedef float v8f __attribute__((ext_vector_type(8)));

#define NB  4096   // blocks in pass 1 (16 KB of d_ws for partials)
#define NT  256    // 8 wave32 per block

// Wave32 sum-reduction using the f32 WMMA unit.
// A (16x4, striped): lanes 0-15 -> A[m][0] (v0), A[m][1] (v1);
//                    lanes 16-31 -> A[m][2] (v0), A[m][3] (v1).
// With v0=acc, v1=0 and B=ones(4x16):  D[m][n] = acc[m] + acc[m+16]  (all n).
// Lane L<16 holds D[0..7][L] in its 8 C-VGPRs, lane L>=16 holds D[8..15][L-16].
// So sum-of-8-VGPRs + shfl_xor(16) == full wave sum in every lane.
__device__ __forceinline__ float wave_reduce_wmma(float acc) {
    v2f a; a[0] = acc;  a[1] = 0.0f;
    v2f b; b[0] = 1.0f; b[1] = 1.0f;
    v8f c = {};
    c = __builtin_amdgcn_wmma_f32_16x16x4_f32(
            /*neg_a=*/false, a, /*neg_b=*/false, b,
            /*c_mod=*/(short)0, c, /*reuse_a=*/false, /*reuse_b=*/false);
    float s = ((c[0] + c[1]) + (c[2] + c[3])) + ((c[4] + c[5]) + (c[6] + c[7]));
    s += __shfl_xor(s, 16, 32);
    return s;  // every lane: full wave32 sum
}

// Full-block (8 wave32) reduction; result valid in wave 0 (incl. thread 0).
__device__ __forceinline__ float block_reduce(float acc) {
    __shared__ float parts[NT / 32];
    const int lane = threadIdx.x & 31;
    const int wid  = threadIdx.x >> 5;

    float w = wave_reduce_wmma(acc);           // EXEC all-1s here: no divergence yet
    if (lane == 0) parts[wid] = w;
    __syncthreads();

    float v = 0.0f;
    if (wid == 0) {
        v = (lane < (NT / 32)) ? parts[lane] : 0.0f;
        v += __shfl_xor(v, 1, 32);
        v += __shfl_xor(v, 2, 32);
        v += __shfl_xor(v, 4, 32);
    }
    return v;
}

// Pass 1: streaming counted loop over float4 pairs; one partial per block.
__global__ void __launch_bounds__(NT)
frob_partial_kernel(const float* __restrict__ o, const float* __restrict__ t,
                    float* __restrict__ partial, long long n4) {
    const v4f* __restrict__ o4 = (const v4f*)o;
    const v4f* __restrict__ t4 = (const v4f*)t;

    const long long stride = (long long)gridDim.x * NT;
    const long long idx    = (long long)blockIdx.x * NT + threadIdx.x;

    // Counted trip count (no per-iteration bounds branch) so the unroller can
    // batch loads: 4x unroll = 8 global_load_b128 in flight per wave.
    const long long niter = (idx < n4) ? ((n4 - idx + stride - 1) / stride) : 0;

    v2f acc = {0.0f, 0.0f};                    // packed accumulator -> v_pk_fma_f32
    long long i = idx;
#pragma unroll 4
    for (long long it = 0; it < niter; ++it, i += stride) {
        v4f a = __builtin_nontemporal_load(o4 + i);   // global_load_b128, TH=NT
        v4f b = __builtin_nontemporal_load(t4 + i);
        v4f d = a - b;                                // 2x v_pk_add_f32 (neg)
        v2f dlo = d.lo, dhi = d.hi;
        acc = dlo * dlo + acc;                        // v_pk_fma_f32
        acc = dhi * dhi + acc;                        // v_pk_fma_f32
    }

    float bsum = block_reduce(acc[0] + acc[1]);
    if (threadIdx.x == 0) partial[blockIdx.x] = bsum;
}

// Pass 2: one block reduces the block partials (+ scalar tail) and takes sqrt.
__global__ void __launch_bounds__(NT)
frob_final_kernel(const float* __restrict__ partial, int nparts,
                  const float* __restrict__ o, const float* __restrict__ t,
                  long long tail_base, int tail_n, float* __restrict__ out) {
    float acc = 0.0f;
    for (int i = threadIdx.x; i < nparts; i += NT) acc += partial[i];
    if ((int)threadIdx.x < tail_n) {                 // n % 4 leftovers (0 here)
        float d = o[tail_base + threadIdx.x] - t[tail_base + threadIdx.x];
        acc = fmaf(d, d, acc);
    }
    float s = block_reduce(acc);
    if (threadIdx.x == 0) out[0] = sqrtf(s);
}

extern "C" void kernel_launch(void* const* d_in, const int* in_sizes, int n_in,
                              void* d_out, int out_size, void* d_ws, size_t ws_size,
                              hipStream_t stream) {
    const float* o = (const float*)d_in[0];
    const float* t = (const float*)d_in[1];
    float* out     = (float*)d_out;
    float* partial = (float*)d_ws;

    const long long n  = (long long)in_sizes[0];   // 4096*8192 = 33,554,432
    const long long n4 = n >> 2;                   // exact float4 count
    const int tail_n   = (int)(n & 3);             // 0 for this shape
    const long long tail_base = n4 << 2;

    int nb = NB;
    if (ws_size < (size_t)nb * sizeof(float))
        nb = (int)(ws_size / sizeof(float));       // defensive; ws is ample in practice
    if (nb < 1) nb = 1;

    frob_partial_kernel<<<nb, NT, 0, stream>>>(o, t, partial, n4);
    frob_final_kernel<<<1, NT, 0, stream>>>(partial, nb, o, t, tail_base, tail_n, out);
}